// Graph_sampler_33732673142785
// MI455X (gfx1250) — compile-verified
//
#include <hip/hip_runtime.h>
#include <math.h>

typedef __attribute__((ext_vector_type(2))) float v2f;
typedef __attribute__((ext_vector_type(8))) float v8f;

#define Bsz   256
#define Nsz   512
#define Msz   16
#define SIGsz 256
#define HIDsz 256
#define Hsz   256

// ---------------------------------------------------------------------------
// K0: per-(b,n) sampling, loc/feat, log-prob reduction.
// field(n,m) == raw[n>>5, j]  (constant in m) => gather is idx-independent.
// ---------------------------------------------------------------------------
__global__ __launch_bounds__(512) void sample_kernel(
    const float* __restrict__ x, const float* __restrict__ z,
    const float* __restrict__ p,
    float* __restrict__ loc, float* __restrict__ feat,
    float* __restrict__ logprob_out)
{
    const int b = blockIdx.x;
    const int n = threadIdx.x;            // 512 threads = 16 waves
    const float* r = p + (n >> 5) * 6;
    const float mux = r[1], muy = r[2];
    const float sx = expf(r[3]), sy = expf(r[4]);
    const float rho = tanhf(r[5]);
    const float omr2 = 1.0f - rho * rho;

    const size_t zi = ((size_t)b * Nsz + n) * 2;
    const float z1 = z[zi], z2 = z[zi + 1];
    const float xs = mux + sx * z1;
    const float ys = muy + sy * (rho * z1 + sqrtf(omr2) * z2);

    const float dx = xs - mux, dy = ys - muy;
    const float quad = (dx * dx / (sx * sx)
                        - 2.0f * rho * dx * dy / (sx * sy)
                        + dy * dy / (sy * sy)) / omr2;
    const float lp = -1.8378770664093453f - logf(sx * sy)
                     - 0.5f * logf(omr2) - 0.5f * quad;

    __shared__ float red[512];
    red[n] = lp;
    __syncthreads();
    for (int st = 256; st > 0; st >>= 1) {
        if (n < st) red[n] += red[n + st];
        __syncthreads();
    }
    if (n == 0) logprob_out[b] = red[0];

    const float lx = tanhf(xs), ly = tanhf(ys);
    loc[zi] = lx; loc[zi + 1] = ly;

    int p0 = (int)truncf(0.5f * (lx + 1.0f) * (float)Hsz - 0.1f);
    int p1 = (int)truncf(0.5f * (ly + 1.0f) * (float)Hsz - 0.1f);
    p0 = min(max(p0, 0), Hsz - 1);
    p1 = min(max(p1, 0), Hsz - 1);
    const float pv = x[(size_t)b * Hsz * Hsz + (size_t)p0 * Hsz + p1];

    const size_t fi = ((size_t)b * Nsz + n) * 3;
    feat[fi] = pv; feat[fi + 1] = lx; feat[fi + 2] = ly;
}

// ---------------------------------------------------------------------------
// K0b: fw[b,n,h] = feat[b,n,:] @ gc1_w  (K=3, trivial)
// ---------------------------------------------------------------------------
__global__ void fw_kernel(const float* __restrict__ feat,
                          const float* __restrict__ w1,
                          float* __restrict__ fw)
{
    const size_t bn = blockIdx.x;
    const int h = threadIdx.x;
    const float f0 = feat[bn * 3], f1 = feat[bn * 3 + 1], f2 = feat[bn * 3 + 2];
    fw[bn * HIDsz + h] = f0 * w1[h] + f1 * w1[HIDsz + h] + f2 * w1[2 * HIDsz + h];
}

// ---------------------------------------------------------------------------
// K1: out[b] = relu(adj(b) @ Bsrc[b] + bias);  adj tiles built on-the-fly in LDS.
// grid = (32 i-tiles, 256 b), block = 512 (16 waves, each owns 16 h-columns).
// ---------------------------------------------------------------------------
__global__ __launch_bounds__(512) void adj_gemm(
    const float* __restrict__ loc, const float* __restrict__ Bsrc,
    const float* __restrict__ bias, float* __restrict__ out)
{
    const int b  = blockIdx.y;
    const int i0 = blockIdx.x * 16;
    const int tid  = threadIdx.x;
    const int lane = tid & 31, wave = tid >> 5;
    const int m  = lane & 15, hl = lane >> 4;
    const int h0 = wave * 16;

    __shared__ float adjT[16 * 16];      // [k][m] — conflict-free frag reads
    __shared__ float lxi[16], lyi[16];

    const float* locb = loc + (size_t)b * (Nsz * 2);
    if (tid < 16) { lxi[tid] = locb[(i0 + tid) * 2]; lyi[tid] = locb[(i0 + tid) * 2 + 1]; }

    const float* Bb = Bsrc + (size_t)b * (Nsz * HIDsz);
    v8f acc0 = {}, acc1 = {};

    for (int kb = 0; kb < Nsz; kb += 16) {
        __syncthreads();                 // guards lxi (iter 0) + adjT reuse
        if (tid < 256) {
            const int mm = tid & 15, kk = tid >> 4;
            const float lxj = locb[(kb + kk) * 2];
            const float lyj = locb[(kb + kk) * 2 + 1];
            const float ddx = lxi[mm] - lxj, ddy = lyi[mm] - lyj;
            adjT[kk * 16 + mm] = 1.0f / (1.0f + 0.5f * (ddx * ddx + ddy * ddy));
        }
        __syncthreads();
#pragma unroll
        for (int s = 0; s < 4; ++s) {
            const int kl = 4 * s + 2 * hl;
            v2f a, bb;
            a.x = adjT[kl * 16 + m];
            a.y = adjT[(kl + 1) * 16 + m];
            const float* bp = Bb + (size_t)(kb + kl) * HIDsz + h0 + m;
            bb.x = bp[0]; bb.y = bp[HIDsz];
            if (s & 1)
                acc1 = __builtin_amdgcn_wmma_f32_16x16x4_f32(false, a, false, bb,
                                                             (short)0, acc1, false, false);
            else
                acc0 = __builtin_amdgcn_wmma_f32_16x16x4_f32(false, a, false, bb,
                                                             (short)0, acc0, false, false);
        }
    }

    const float bv = bias[h0 + m];
    float* ob = out + (size_t)b * (Nsz * HIDsz);
#pragma unroll
    for (int r2 = 0; r2 < 8; ++r2) {
        const int row = r2 + 8 * hl;
        const float v = acc0[r2] + acc1[r2] + bv;
        ob[(size_t)(i0 + row) * HIDsz + h0 + m] = fmaxf(v, 0.0f);
    }
}

// ---------------------------------------------------------------------------
// K2/K3: generic C[16 x 256] = A[16 x K] @ B[K x 256] (row-major, ldb=ldc=256).
// A tile staged transposed in LDS (pad 17). block = 512 (16 waves = full 256 cols).
// ---------------------------------------------------------------------------
__global__ __launch_bounds__(512) void gemm16(
    const float* __restrict__ A, long long aBSX, long long aBSY, int lda,
    const float* __restrict__ Bm, long long bSY,
    float* __restrict__ C, long long cBSX, long long cSY,
    int K, int klog)
{
    __shared__ float As[512 * 17];       // supports K up to 512

    const int tid  = threadIdx.x;
    const int lane = tid & 31, wave = tid >> 5;
    const int m  = lane & 15, hl = lane >> 4;
    const int h0 = wave * 16;

    const float* Ab = A + (long long)blockIdx.x * aBSX + (long long)blockIdx.y * aBSY;
    const float* Bb = Bm + (long long)blockIdx.y * bSY;
    float*       Cb = C + (long long)blockIdx.x * cBSX + (long long)blockIdx.y * cSY;

    for (int idx = tid; idx < (K << 4); idx += 512) {
        const int r = idx >> klog;       // row 0..15
        const int c = idx & (K - 1);     // col 0..K-1 (K is power of two)
        As[c * 17 + r] = Ab[(size_t)r * lda + c];
    }
    __syncthreads();

    v8f acc0 = {}, acc1 = {};
    for (int kb = 0; kb < K; kb += 16) {
#pragma unroll
        for (int s = 0; s < 4; ++s) {
            const int k0 = kb + 4 * s + 2 * hl;
            v2f a, bb;
            a.x = As[k0 * 17 + m];
            a.y = As[(k0 + 1) * 17 + m];
            const float* bp = Bb + (size_t)k0 * 256 + h0 + m;
            bb.x = bp[0]; bb.y = bp[256];
            if (s & 1)
                acc1 = __builtin_amdgcn_wmma_f32_16x16x4_f32(false, a, false, bb,
                                                             (short)0, acc1, false, false);
            else
                acc0 = __builtin_amdgcn_wmma_f32_16x16x4_f32(false, a, false, bb,
                                                             (short)0, acc0, false, false);
        }
    }
#pragma unroll
    for (int r2 = 0; r2 < 8; ++r2) {
        const int row = r2 + 8 * hl;
        Cb[(size_t)row * 256 + h0 + m] = acc0[r2] + acc1[r2];
    }
}

// ---------------------------------------------------------------------------
// K4: online softmax reduction over batch axis. l is [b][s][h]; thread = (s,h).
// Lane-contiguous (s,h) => coalesced reads at every b.
// ---------------------------------------------------------------------------
__global__ void softmax_reduce(const float* __restrict__ l,
                               float* __restrict__ mx, float* __restrict__ dn)
{
    const int sh = blockIdx.x * blockDim.x + threadIdx.x;   // 0..65535
    float mcur = -3.402823466e38f, d = 0.0f;
    for (int b = 0; b < Bsz; ++b) {
        const float v = l[(size_t)b * (SIGsz * HIDsz) + sh];
        const float mn = fmaxf(mcur, v);
        d = d * expf(mcur - mn) + expf(v - mn);
        mcur = mn;
    }
    mx[sh] = mcur; dn[sh] = d;
}

// ---------------------------------------------------------------------------
// K5: sig[b,s] = sum_h exp(l1-m1)/d1 + exp(l2-m2)/d2. One wave per (b,s).
// ---------------------------------------------------------------------------
__global__ void sig_kernel(const float* __restrict__ l1, const float* __restrict__ l2,
                           const float* __restrict__ m1, const float* __restrict__ d1,
                           const float* __restrict__ m2, const float* __restrict__ d2,
                           float* __restrict__ sig)
{
    const int p = blockIdx.x * 8 + (threadIdx.x >> 5);      // p = b*256 + s
    const int lane = threadIdx.x & 31;
    const int s = p & 255;
    const size_t base = (size_t)p * HIDsz;
    const int shb = s * HIDsz;
    float acc = 0.0f;
#pragma unroll
    for (int j = 0; j < 8; ++j) {
        const int h = lane + j * 32;
        acc += expf(l1[base + h] - m1[shb + h]) / d1[shb + h]
             + expf(l2[base + h] - m2[shb + h]) / d2[shb + h];
    }
    for (int off = 16; off; off >>= 1) acc += __shfl_down(acc, off, 32);
    if (lane == 0) sig[p] = acc;
}

// ---------------------------------------------------------------------------
// K6: out[b,:] = softmax(sig[b] @ fcf_w^T + fcf_b). One wave per b.
// ---------------------------------------------------------------------------
__global__ void out_kernel(const float* __restrict__ sig,
                           const float* __restrict__ fw_, const float* __restrict__ fb,
                           float* __restrict__ outp)
{
    const int b = blockIdx.x;
    const int lane = threadIdx.x;
    float acc[10];
#pragma unroll
    for (int o = 0; o < 10; ++o) acc[o] = 0.0f;
    for (int s = lane; s < SIGsz; s += 32) {
        const float v = sig[b * SIGsz + s];
#pragma unroll
        for (int o = 0; o < 10; ++o) acc[o] += v * fw_[o * SIGsz + s];
    }
#pragma unroll
    for (int o = 0; o < 10; ++o)
        for (int off = 16; off; off >>= 1) acc[o] += __shfl_down(acc[o], off, 32);
    if (lane == 0) {
        float lg[10], mxv = -3.402823466e38f;
#pragma unroll
        for (int o = 0; o < 10; ++o) { lg[o] = acc[o] + fb[o]; mxv = fmaxf(mxv, lg[o]); }
        float sm = 0.0f;
#pragma unroll
        for (int o = 0; o < 10; ++o) { lg[o] = expf(lg[o] - mxv); sm += lg[o]; }
#pragma unroll
        for (int o = 0; o < 10; ++o) outp[b * 10 + o] = lg[o] / sm;
    }
}

// ---------------------------------------------------------------------------
extern "C" void kernel_launch(void* const* d_in, const int* in_sizes, int n_in,
                              void* d_out, int out_size, void* d_ws, size_t ws_size,
                              hipStream_t stream)
{
    (void)in_sizes; (void)n_in; (void)out_size; (void)ws_size;
    const float* x      = (const float*)d_in[0];
    // d_in[1] = unif : provably unused (mixture fields constant across m)
    const float* z      = (const float*)d_in[2];
    const float* mg     = (const float*)d_in[3];
    const float* gc1_w  = (const float*)d_in[4];
    const float* gc1_b  = (const float*)d_in[5];
    const float* gc2_w  = (const float*)d_in[6];
    const float* gc2_b  = (const float*)d_in[7];
    const float* c1     = (const float*)d_in[8];
    const float* c2     = (const float*)d_in[9];
    const float* fcf_w  = (const float*)d_in[10];
    const float* fcf_b  = (const float*)d_in[11];
    float* outp = (float*)d_out;                 // [0,2560) out, [2560,2816) logprob

    // ---- workspace carve (aligned 256B) ----
    char* w = (char*)d_ws;
    size_t off = 0;
    auto carve = [&](size_t bytes) -> float* {
        float* ptr = (float*)(w + off);
        off += (bytes + 255) & ~(size_t)255;
        return ptr;
    };
    float* loc  = carve((size_t)Bsz * Nsz * 2 * 4);       // 1 MiB
    float* feat = carve((size_t)Bsz * Nsz * 3 * 4);       // 1.5 MiB
    float* m1   = carve((size_t)SIGsz * HIDsz * 4);
    float* d1   = carve((size_t)SIGsz * HIDsz * 4);
    float* m2   = carve((size_t)SIGsz * HIDsz * 4);
    float* d2   = carve((size_t)SIGsz * HIDsz * 4);
    float* sig  = carve((size_t)Bsz * SIGsz * 4);
    float* buf1 = carve((size_t)Bsz * Nsz * HIDsz * 4);   // fw -> t -> l1|l2
    float* s1   = carve((size_t)Bsz * Nsz * HIDsz * 4);
    float* s2   = carve((size_t)Bsz * Nsz * HIDsz * 4);
    float* l1 = buf1;
    float* l2 = buf1 + (size_t)Bsz * SIGsz * HIDsz;       // second 64 MiB half

    // K0: sampling + loc/feat + full_log_prob
    sample_kernel<<<Bsz, 512, 0, stream>>>(x, z, mg, loc, feat, outp + Bsz * 10);
    // K0b: fw = feat @ gc1_w
    fw_kernel<<<Bsz * Nsz, HIDsz, 0, stream>>>(feat, gc1_w, buf1);
    // K1: s1 = relu(adj @ fw + gc1_b)
    adj_gemm<<<dim3(Nsz / 16, Bsz), 512, 0, stream>>>(loc, buf1, gc1_b, s1);
    // K2: t = s1 @ gc2_w  (flattened (B*N,256)x(256,256)); t overwrites fw
    gemm16<<<dim3(Bsz * Nsz / 16, 1), 512, 0, stream>>>(
        s1, 16LL * 256, 0LL, 256, gc2_w, 0LL,
        buf1, 16LL * 256, 0LL, 256, 8);
    // K1': s2 = relu(adj @ t + gc2_b)
    adj_gemm<<<dim3(Nsz / 16, Bsz), 512, 0, stream>>>(loc, buf1, gc2_b, s2);
    // K3: l1[b] = c1 @ s1[b];  l2[b] = c2 @ s2[b]   (overwrite buf1)
    gemm16<<<dim3(SIGsz / 16, Bsz), 512, 0, stream>>>(
        c1, 16LL * 512, 0LL, 512, s1, (long long)Nsz * HIDsz,
        l1, 16LL * 256, (long long)SIGsz * HIDsz, 512, 9);
    gemm16<<<dim3(SIGsz / 16, Bsz), 512, 0, stream>>>(
        c2, 16LL * 512, 0LL, 512, s2, (long long)Nsz * HIDsz,
        l2, 16LL * 256, (long long)SIGsz * HIDsz, 512, 9);
    // K4: batch-axis softmax stats
    softmax_reduce<<<SIGsz * HIDsz / 256, 256, 0, stream>>>(l1, m1, d1);
    softmax_reduce<<<SIGsz * HIDsz / 256, 256, 0, stream>>>(l2, m2, d2);
    // K5: sig
    sig_kernel<<<Bsz * SIGsz / 8, 256, 0, stream>>>(l1, l2, m1, d1, m2, d2, sig);
    // K6: final softmax
    out_kernel<<<Bsz, 32, 0, stream>>>(sig, fcf_w, fcf_b, outp);
}